// LinearAttention_79276506350102
// MI455X (gfx1250) — compile-verified
//
#include <hip/hip_runtime.h>
#include <math.h>
#include <stdint.h>

typedef float v2f __attribute__((ext_vector_type(2)));
typedef float v8f __attribute__((ext_vector_type(8)));

#define NTOK   131072        // B*L = 16*8192
#define SEQ    8192
#define NB     16
#define HEADS  4
#define DHEAD  32
#define HID    128           // HEADS*DHEAD
#define DIN    256
#define NQKV   384

// --------------------------------------------------------------------------
// CDNA5 async global->LDS copy (16B per lane), tracked by ASYNCcnt.
// --------------------------------------------------------------------------
__device__ __forceinline__ uint32_t lds_off(const void* p) {
    // flat shared pointer: addr[31:0] is the LDS byte offset
    return (uint32_t)(uintptr_t)p;
}
__device__ __forceinline__ void async_cp16(uint32_t lds_byte, const float* gsrc) {
    asm volatile("global_load_async_to_lds_b128 %0, %1, off"
                 :: "v"(lds_byte), "v"(gsrc)
                 : "memory");
}
__device__ __forceinline__ void wait_async_le4() {
    asm volatile("s_wait_asynccnt 0x4" ::: "memory");
}
__device__ __forceinline__ void wait_async_0() {
    asm volatile("s_wait_asynccnt 0x0" ::: "memory");
}

// ---------------------------------------------------------------------------
// Kernel 1: qkv = x @ W_qkv.   Block tile: 64 rows x 128 cols, 8 waves.
// Wave w -> 16 cols, 4 row-tiles of 16.  A double-buffered in LDS via
// async-to-LDS, K chunked by 64.
// strip 0 -> q (scaled), strip 1 -> k, strip 2 -> v, each stored [NTOK,128].
// ---------------------------------------------------------------------------
__global__ __launch_bounds__(256) void qkv_gemm(const float* __restrict__ x,
                                                const float* __restrict__ Wqkv,
                                                float* __restrict__ qs,
                                                float* __restrict__ kbuf,
                                                float* __restrict__ vbuf)
{
    __shared__ float As[2][64][68];              // 68-stride: conflict-free column reads
    const int tid   = threadIdx.x;
    const int wave  = tid >> 5;
    const int lane  = tid & 31;
    const int row0  = blockIdx.x * 64;
    const int strip = blockIdx.y;                // 0=q 1=k 2=v
    const int n0    = strip * 128 + wave * 16;   // global output column base
    const int mloc  = lane & 15;
    const int kb2   = (lane >> 4) << 1;          // 0 or 2 (K sub-index)

    v8f acc[4] = {};

    auto stage = [&](int buf, int k0) {
        #pragma unroll
        for (int i = 0; i < 4; ++i) {
            int cidx = tid + i * 256;            // 0..1023 : 16B chunk index
            int r  = cidx >> 4;                  // row 0..63
            int c4 = (cidx & 15) << 2;           // col 0,4,...,60
            async_cp16(lds_off(&As[buf][r][c4]),
                       &x[(size_t)(row0 + r) * DIN + (k0 + c4)]);
        }
    };

    stage(0, 0);
    for (int c = 0; c < 4; ++c) {                // 4 K-chunks of 64
        const int k0 = c * 64;
        if (c + 1 < 4) { stage((c + 1) & 1, k0 + 64); wait_async_le4(); }
        else           { wait_async_0(); }
        __syncthreads();
        const int cur = c & 1;
        #pragma unroll
        for (int kk = 0; kk < 64; kk += 4) {
            const float* bp = &Wqkv[(size_t)(k0 + kk + kb2) * NQKV + (n0 + mloc)];
            v2f bfrag;
            bfrag[0] = bp[0];
            bfrag[1] = bp[NQKV];
            #pragma unroll
            for (int mt = 0; mt < 4; ++mt) {
                v2f afrag = *(const v2f*)&As[cur][mt * 16 + mloc][kk + kb2];
                acc[mt] = __builtin_amdgcn_wmma_f32_16x16x4_f32(
                    false, afrag, false, bfrag, (short)0, acc[mt], false, false);
            }
        }
        __syncthreads();
    }

    const float scale = 0.17677669529663687f;    // 32^-0.5
    float* dst = (strip == 0) ? qs : (strip == 1) ? kbuf : vbuf;
    const float mul = (strip == 0) ? scale : 1.0f;
    const int coll = wave * 16 + mloc;           // column inside the 128-wide plane
    const int mhi  = (lane >> 4) << 3;           // 0 or 8
    #pragma unroll
    for (int mt = 0; mt < 4; ++mt) {
        #pragma unroll
        for (int r = 0; r < 8; ++r) {
            int row = row0 + mt * 16 + r + mhi;
            dst[(size_t)row * HID + coll] = acc[mt][r] * mul;
        }
    }
}

// ---------------------------------------------------------------------------
// Kernel 2: per-(batch, channel) softmax stats over L: max and 1/sum(exp).
// ---------------------------------------------------------------------------
__global__ __launch_bounds__(256) void kstats(const float* __restrict__ kbuf,
                                              float* __restrict__ mArr,
                                              float* __restrict__ invsArr)
{
    const int bc  = blockIdx.x;                  // 0..2047
    const int b   = bc >> 7;
    const int col = bc & 127;
    const float* base = kbuf + (size_t)b * SEQ * HID + col;
    __shared__ float red[256];

    float lm = -INFINITY;
    for (int l = threadIdx.x; l < SEQ; l += 256)
        lm = fmaxf(lm, base[(size_t)l * HID]);
    red[threadIdx.x] = lm;
    __syncthreads();
    for (int s = 128; s > 0; s >>= 1) {
        if (threadIdx.x < s) red[threadIdx.x] = fmaxf(red[threadIdx.x], red[threadIdx.x + s]);
        __syncthreads();
    }
    const float mx = red[0];
    __syncthreads();

    float ls = 0.0f;
    for (int l = threadIdx.x; l < SEQ; l += 256)
        ls += expf(base[(size_t)l * HID] - mx);
    red[threadIdx.x] = ls;
    __syncthreads();
    for (int s = 128; s > 0; s >>= 1) {
        if (threadIdx.x < s) red[threadIdx.x] += red[threadIdx.x + s];
        __syncthreads();
    }
    if (threadIdx.x == 0) {
        mArr[bc]    = mx;
        invsArr[bc] = 1.0f / red[0];
    }
}

// ---------------------------------------------------------------------------
// Kernel 3: context[b,h] = softmax(k)^T @ v  (32x32 per (b,h)).
// One block of 1024 threads per (b,h); thread = (d,e).
// ---------------------------------------------------------------------------
__global__ __launch_bounds__(1024) void context_k(const float* __restrict__ kbuf,
                                                  const float* __restrict__ vbuf,
                                                  const float* __restrict__ mArr,
                                                  const float* __restrict__ invsArr,
                                                  float* __restrict__ Cm)
{
    const int b = blockIdx.x >> 2;
    const int h = blockIdx.x & 3;
    const int t = threadIdx.x;
    const int d = t >> 5;
    const int e = t & 31;

    const float md = mArr[b * HID + h * DHEAD + d];
    const float is = invsArr[b * HID + h * DHEAD + d];

    __shared__ float ks[32][33];
    __shared__ float ps[32][33];
    __shared__ float vs[32][32];

    float acc = 0.0f;
    const size_t rowbase = (size_t)b * SEQ;
    for (int l0 = 0; l0 < SEQ; l0 += 32) {
        // cooperative load of 32 rows x 32 channels of k and v for this head
        ks[d][e] = kbuf[(rowbase + l0 + d) * HID + h * DHEAD + e];
        vs[d][e] = vbuf[(rowbase + l0 + d) * HID + h * DHEAD + e];
        __syncthreads();
        // thread t computes probability for (row = e, channel = d)
        ps[e][d] = expf(ks[e][d] - md) * is;
        __syncthreads();
        #pragma unroll
        for (int i = 0; i < 32; ++i)
            acc += ps[i][d] * vs[i][e];
        __syncthreads();
    }
    Cm[(((size_t)b * HEADS + h) * DHEAD + d) * DHEAD + e] = acc;
}

// ---------------------------------------------------------------------------
// Kernel 4: fold context into W_out:  Wc[b][h*32+d][j] = sum_e C[b,h,d,e]*W_out[h*32+e][j]
// ---------------------------------------------------------------------------
__global__ __launch_bounds__(256) void fold_k(const float* __restrict__ Cm,
                                              const float* __restrict__ Wout,
                                              float* __restrict__ Wc)
{
    const int blk = blockIdx.x;                  // b*128 + f
    const int b = blk >> 7;
    const int f = blk & 127;
    const int h = f >> 5;
    const int d = f & 31;
    const int j = threadIdx.x;                   // 0..255

    __shared__ float crow[32];
    if (j < 32) crow[j] = Cm[(((size_t)b * HEADS + h) * DHEAD + d) * DHEAD + j];
    __syncthreads();

    float acc = 0.0f;
    #pragma unroll
    for (int e = 0; e < 32; ++e)
        acc += crow[e] * Wout[(h * DHEAD + e) * DIN + j];
    Wc[((size_t)b * HID + f) * DIN + j] = acc;
}

// ---------------------------------------------------------------------------
// Kernel 5: y[b] = q_scaled[b] @ Wc[b] + b_out.   Same WMMA tiling as kernel 1,
// double-buffered async-to-LDS staging (2 K-chunks).
// ---------------------------------------------------------------------------
__global__ __launch_bounds__(256) void out_gemm(const float* __restrict__ qs,
                                                const float* __restrict__ Wc,
                                                const float* __restrict__ bout,
                                                float* __restrict__ y)
{
    __shared__ float As[2][64][68];
    const int tid  = threadIdx.x;
    const int wave = tid >> 5;
    const int lane = tid & 31;
    const int row0 = blockIdx.x * 64;
    const int b    = row0 >> 13;                 // 8192 rows per batch
    const int n0   = blockIdx.y * 128 + wave * 16;
    const int mloc = lane & 15;
    const int kb2  = (lane >> 4) << 1;

    v8f acc[4] = {};

    auto stage = [&](int buf, int k0) {
        #pragma unroll
        for (int i = 0; i < 4; ++i) {
            int cidx = tid + i * 256;
            int r  = cidx >> 4;
            int c4 = (cidx & 15) << 2;
            async_cp16(lds_off(&As[buf][r][c4]),
                       &qs[(size_t)(row0 + r) * HID + (k0 + c4)]);
        }
    };

    stage(0, 0);
    for (int c = 0; c < 2; ++c) {                // 2 K-chunks of 64
        const int k0 = c * 64;
        if (c + 1 < 2) { stage((c + 1) & 1, k0 + 64); wait_async_le4(); }
        else           { wait_async_0(); }
        __syncthreads();
        const int cur = c & 1;
        #pragma unroll
        for (int kk = 0; kk < 64; kk += 4) {
            const float* bp = &Wc[((size_t)b * HID + (k0 + kk + kb2)) * DIN + (n0 + mloc)];
            v2f bfrag;
            bfrag[0] = bp[0];
            bfrag[1] = bp[DIN];
            #pragma unroll
            for (int mt = 0; mt < 4; ++mt) {
                v2f afrag = *(const v2f*)&As[cur][mt * 16 + mloc][kk + kb2];
                acc[mt] = __builtin_amdgcn_wmma_f32_16x16x4_f32(
                    false, afrag, false, bfrag, (short)0, acc[mt], false, false);
            }
        }
        __syncthreads();
    }

    const int col = n0 + mloc;
    const float bias = bout[col];
    const int mhi = (lane >> 4) << 3;
    #pragma unroll
    for (int mt = 0; mt < 4; ++mt) {
        #pragma unroll
        for (int r = 0; r < 8; ++r) {
            int row = row0 + mt * 16 + r + mhi;
            y[(size_t)row * DIN + col] = acc[mt][r] + bias;
        }
    }
}

// ---------------------------------------------------------------------------
extern "C" void kernel_launch(void* const* d_in, const int* in_sizes, int n_in,
                              void* d_out, int out_size, void* d_ws, size_t ws_size,
                              hipStream_t stream)
{
    (void)in_sizes; (void)n_in; (void)out_size; (void)ws_size;
    const float* x    = (const float*)d_in[0];
    const float* Wqkv = (const float*)d_in[1];
    const float* Wout = (const float*)d_in[2];
    const float* bout = (const float*)d_in[3];
    float* y  = (float*)d_out;
    float* ws = (float*)d_ws;

    const size_t plane = (size_t)NTOK * HID;     // 16.7M floats each
    float* qs      = ws;
    float* kbuf    = qs + plane;
    float* vbuf    = kbuf + plane;
    float* mArr    = vbuf + plane;               // 2048
    float* invsArr = mArr + (size_t)NB * HID;    // 2048
    float* Cm      = invsArr + (size_t)NB * HID; // 65536
    float* Wc      = Cm + (size_t)NB * HEADS * DHEAD * DHEAD;  // 524288

    qkv_gemm<<<dim3(NTOK / 64, 3), 256, 0, stream>>>(x, Wqkv, qs, kbuf, vbuf);
    kstats<<<NB * HID, 256, 0, stream>>>(kbuf, mArr, invsArr);
    context_k<<<NB * HEADS, 1024, 0, stream>>>(kbuf, vbuf, mArr, invsArr, Cm);
    fold_k<<<NB * HID, 256, 0, stream>>>(Cm, Wout, Wc);
    out_gemm<<<dim3(NTOK / 64, 2), 256, 0, stream>>>(qs, Wc, bout, y);
}